// LookForTheChangeLoss_61838939128289
// MI455X (gfx1250) — compile-verified
//
#include <hip/hip_runtime.h>
#include <hip/hip_bf16.h>
#include <math.h>
#include <stdint.h>
#include <limits.h>

#define NTHREADS 256
#define TLEN     4096
#define EPT      (TLEN / NTHREADS)   // 16 elements per thread (contiguous chunk)
#define DELTA_   16
#define KAPPA_   4
#define NEGV     (-1e30f)
#define ACTION_LOSS_WEIGHT 0.2f

__global__ __launch_bounds__(NTHREADS)
void lftc_loss_kernel(const float* __restrict__ s12,   // [B, T, 2]
                      const float* __restrict__ act,   // [B, T]
                      const int*   __restrict__ lens,  // [B]
                      const float* __restrict__ vscore,// [B]
                      float* __restrict__ partial)     // [B, 2]
{
  __shared__ float s_pre[TLEN];        // cummax of masked p1
  __shared__ float s_suf[TLEN];        // suffix cummax of masked p2
  __shared__ float s_zp2[TLEN];        // async-DMA'd z row, later reused as p2m originals
  __shared__ float s_chunk[NTHREADS];  // scan scratch
  __shared__ float s_redf[NTHREADS];   // reduction scratch (float)
  __shared__ int   s_redi[NTHREADS];   // reduction scratch (int)

  const int   b    = blockIdx.x;
  const int   tid  = threadIdx.x;
  const int   base = tid * EPT;
  const int   len  = lens[b];
  const float vs   = vscore[b];

  // ---------------------------------------------------------------
  // Phase 0: async DMA of this row of action logits (16 KB) into LDS
  //          via gfx1250 GLOBAL_LOAD_ASYNC_TO_LDS_B128 (ASYNCcnt).
  //          Overlaps with the softmax+scan work on the s1s2 row.
  // ---------------------------------------------------------------
  {
    const float* gsrc  = act + (size_t)b * TLEN + base;
    uint32_t ldsoff    = (uint32_t)(uintptr_t)&s_zp2[base];
    uint64_t gaddr     = (uint64_t)(uintptr_t)gsrc;
    #pragma unroll
    for (int k = 0; k < EPT / 4; ++k) {
      asm volatile("global_load_async_to_lds_b128 %0, %1, off"
                   :: "v"(ldsoff + 16u * (unsigned)k), "v"(gaddr + 16u * (unsigned)k)
                   : "memory");
    }
  }

  // ---------------------------------------------------------------
  // Phase 1: load s1s2 chunk (B128), softmax(2), mask, local cummax
  // ---------------------------------------------------------------
  float p2v[EPT];
  {
    const float4* rowp = (const float4*)(s12 + ((size_t)b * TLEN + base) * 2);
    float run = NEGV;
    #pragma unroll
    for (int q = 0; q < EPT / 2; ++q) {
      float4 v4 = rowp[q];
      // timestep base + 2q : logits (v4.x, v4.y)
      {
        int t = base + 2 * q;
        bool valid = t < len;
        float p1 = 1.0f / (1.0f + expf(v4.y - v4.x));
        float p2 = 1.0f / (1.0f + expf(v4.x - v4.y));
        float p1m = valid ? p1 : NEGV;
        p2v[2 * q] = valid ? p2 : NEGV;
        run = fmaxf(run, p1m);
        s_pre[t] = run;
      }
      // timestep base + 2q + 1 : logits (v4.z, v4.w)
      {
        int t = base + 2 * q + 1;
        bool valid = t < len;
        float p1 = 1.0f / (1.0f + expf(v4.w - v4.z));
        float p2 = 1.0f / (1.0f + expf(v4.z - v4.w));
        float p1m = valid ? p1 : NEGV;
        p2v[2 * q + 1] = valid ? p2 : NEGV;
        run = fmaxf(run, p1m);
        s_pre[t] = run;
      }
    }
    s_chunk[tid] = run;
  }
  __syncthreads();

  // forward Hillis-Steele max-scan over chunk maxima
  for (int s = 1; s < NTHREADS; s <<= 1) {
    float v = s_chunk[tid];
    float u = (tid >= s) ? s_chunk[tid - s] : NEGV;
    __syncthreads();
    s_chunk[tid] = fmaxf(v, u);
    __syncthreads();
  }
  {
    float off = (tid > 0) ? s_chunk[tid - 1] : NEGV;
    #pragma unroll
    for (int j = 0; j < EPT; ++j)
      s_pre[base + j] = fmaxf(s_pre[base + j], off);
  }
  __syncthreads();

  // ---------------------------------------------------------------
  // Phase 2: suffix cummax of masked p2
  // ---------------------------------------------------------------
  {
    float run = NEGV;
    #pragma unroll
    for (int j = EPT - 1; j >= 0; --j) {
      run = fmaxf(run, p2v[j]);
      s_suf[base + j] = run;
    }
    s_chunk[tid] = run;
  }
  __syncthreads();
  for (int s = 1; s < NTHREADS; s <<= 1) {
    float v = s_chunk[tid];
    float u = (tid + s < NTHREADS) ? s_chunk[tid + s] : NEGV;
    __syncthreads();
    s_chunk[tid] = fmaxf(v, u);
    __syncthreads();
  }
  {
    float off = (tid < NTHREADS - 1) ? s_chunk[tid + 1] : NEGV;
    #pragma unroll
    for (int j = 0; j < EPT; ++j)
      s_suf[base + j] = fmaxf(s_suf[base + j], off);
  }

  // ---------------------------------------------------------------
  // Phase 3: drain async DMA, grab z into registers, reuse buffer
  //          to hold the original masked p2 values (for s2 argmax).
  // ---------------------------------------------------------------
  asm volatile("s_wait_asynccnt 0" ::: "memory");
  float zv[EPT];
  #pragma unroll
  for (int j = 0; j < EPT; ++j) zv[j] = s_zp2[base + j];
  #pragma unroll
  for (int j = 0; j < EPT; ++j) s_zp2[base + j] = p2v[j];
  __syncthreads();  // s_pre, s_suf, p2m now visible to all threads

  // ---------------------------------------------------------------
  // Phase 4: ac = argmax_t (first index) of pre[t-D]*sigmoid(z)*suf[t+D]
  // ---------------------------------------------------------------
  float best = NEGV;
  int   bidx = INT_MAX;
  #pragma unroll
  for (int j = 0; j < EPT; ++j) {
    int t = base + j;
    float ps = (t >= DELTA_) ? s_pre[t - DELTA_] : NEGV;
    float ss = (t + DELTA_ < TLEN) ? s_suf[t + DELTA_] : NEGV;
    bool ok = (ps > 0.5f * NEGV) && (ss > 0.5f * NEGV) && (t < len);
    float pa = 1.0f / (1.0f + expf(-zv[j]));
    float sc = ok ? ps * pa * ss : NEGV;
    if (sc > best || (sc == best && t < bidx)) { best = sc; bidx = t; }
  }
  s_redf[tid] = best; s_redi[tid] = bidx;
  __syncthreads();
  for (int s = NTHREADS / 2; s > 0; s >>= 1) {
    if (tid < s) {
      float fo = s_redf[tid + s]; int io = s_redi[tid + s];
      if (fo > s_redf[tid] || (fo == s_redf[tid] && io < s_redi[tid])) {
        s_redf[tid] = fo; s_redi[tid] = io;
      }
    }
    __syncthreads();
  }
  const int ac = s_redi[0];
  __syncthreads();

  // ---------------------------------------------------------------
  // Phase 5: s1 = first t<=ac-D with pre[t]==pre[ac-D]  (cummax trick)
  // ---------------------------------------------------------------
  int m1 = INT_MAX;
  {
    int K = ac - DELTA_;
    if (K >= 0) {
      float M = s_pre[K];
      #pragma unroll
      for (int j = 0; j < EPT; ++j) {
        int t = base + j;
        if (t <= K && s_pre[t] == M && t < m1) m1 = t;
      }
    }
  }
  s_redi[tid] = m1;
  __syncthreads();
  for (int s = NTHREADS / 2; s > 0; s >>= 1) {
    if (tid < s) {
      int o = s_redi[tid + s];
      if (o < s_redi[tid]) s_redi[tid] = o;
    }
    __syncthreads();
  }
  int s1i = s_redi[0]; if (s1i == INT_MAX) s1i = 0;
  __syncthreads();

  // ---------------------------------------------------------------
  // Phase 6: s2 = first t>=ac+D with p2m[t]==suf[ac+D]
  // ---------------------------------------------------------------
  int m2 = INT_MAX;
  {
    int K2 = ac + DELTA_;
    if (K2 < TLEN) {
      float M2 = s_suf[K2];
      #pragma unroll
      for (int j = 0; j < EPT; ++j) {
        int t = base + j;
        if (t >= K2 && s_zp2[t] == M2 && t < m2) m2 = t;
      }
    }
  }
  s_redi[tid] = m2;
  __syncthreads();
  for (int s = NTHREADS / 2; s > 0; s >>= 1) {
    if (tid < s) {
      int o = s_redi[tid + s];
      if (o < s_redi[tid]) s_redi[tid] = o;
    }
    __syncthreads();
  }
  int s2i = s_redi[0]; if (s2i == INT_MAX) s2i = 0;
  __syncthreads();

  // ---------------------------------------------------------------
  // Phase 7: weighted BCE action-loss partial sum (per block)
  // ---------------------------------------------------------------
  float acc = 0.0f;
  #pragma unroll
  for (int j = 0; j < EPT; ++j) {
    int t = base + j;
    if (t < len) {
      float z = zv[j];
      int d = t - ac; d = (d < 0) ? -d : d;
      float y = (d <= KAPPA_) ? 1.0f : 0.0f;
      float w = (d <= KAPPA_) ? 10.0f : 1.0f;
      float bce = fmaxf(z, 0.0f) - z * y + log1pf(expf(-fabsf(z)));
      acc += w * bce;
    }
  }
  s_redf[tid] = acc;
  __syncthreads();
  for (int s = NTHREADS / 2; s > 0; s >>= 1) {
    if (tid < s) s_redf[tid] += s_redf[tid + s];
    __syncthreads();
  }

  // ---------------------------------------------------------------
  // Phase 8: state CE at s1 (class 0) and s2 (class 1); write partials
  // ---------------------------------------------------------------
  if (tid == 0) {
    const float* r1 = s12 + ((size_t)b * TLEN + s1i) * 2;
    float a0 = r1[0], a1 = r1[1];
    float mm = fmaxf(a0, a1);
    float lse = mm + log1pf(expf(fminf(a0, a1) - mm));
    float st = -(a0 - lse);

    const float* r2 = s12 + ((size_t)b * TLEN + s2i) * 2;
    float b0 = r2[0], b1 = r2[1];
    float m2f = fmaxf(b0, b1);
    float lse2 = m2f + log1pf(expf(fminf(b0, b1) - m2f));
    st += -(b1 - lse2);

    partial[2 * b + 0] = st * vs;
    partial[2 * b + 1] = s_redf[0] * vs;
  }
}

__global__ __launch_bounds__(256)
void lftc_finalize_kernel(const float* __restrict__ partial,
                          float* __restrict__ out, int B)
{
  __shared__ float ss[256], sa[256];
  int tid = threadIdx.x;
  float a = 0.0f, c = 0.0f;
  for (int i = tid; i < B; i += 256) {
    a += partial[2 * i + 0];
    c += partial[2 * i + 1];
  }
  ss[tid] = a; sa[tid] = c;
  __syncthreads();
  for (int s = 128; s > 0; s >>= 1) {
    if (tid < s) { ss[tid] += ss[tid + s]; sa[tid] += sa[tid + s]; }
    __syncthreads();
  }
  if (tid == 0) {
    out[0] = ss[0];
    out[1] = ACTION_LOSS_WEIGHT * sa[0];
  }
}

extern "C" void kernel_launch(void* const* d_in, const int* in_sizes, int n_in,
                              void* d_out, int out_size, void* d_ws, size_t ws_size,
                              hipStream_t stream) {
  const float* s12  = (const float*)d_in[0];  // [B,T,2] f32
  const float* act  = (const float*)d_in[1];  // [B,T,1] f32
  const int*   lens = (const int*)d_in[2];    // [B] i32
  const float* vs   = (const float*)d_in[3];  // [B] f32
  const int B = in_sizes[2];

  float* partial = (float*)d_ws;              // [B,2] f32 scratch

  lftc_loss_kernel<<<B, NTHREADS, 0, stream>>>(s12, act, lens, vs, partial);
  lftc_finalize_kernel<<<1, 256, 0, stream>>>(partial, (float*)d_out, B);
}